// GATLayer_50053548868137
// MI455X (gfx1250) — compile-verified
//
#include <hip/hip_runtime.h>

#define E_EDGES   250000
#define K_IN      128
#define N_OUT     256
#define H_HEADS   8
#define NSEG      50000
#define NEG_SLOPE 0.2f

typedef float v2f __attribute__((ext_vector_type(2)));
typedef float v8f __attribute__((ext_vector_type(8)));

// ---------------------------------------------------------------------------
// Kernel 1: proj = messages @ W.T via V_WMMA_F32_16X16X4_F32.
// Block = 256 thr (8 waves). Block tile = 32 rows x 256 cols.
// Wave w owns head w (cols [32w, 32w+32) = two 16-wide N tiles) and both
// 16-row M tiles. Message tile staged in LDS (stride 132 to avoid bank
// conflicts). Tgt-side tiles are stored as the "placeholder" output half;
// per-head attention scores are reduced in the epilogue.
// ---------------------------------------------------------------------------
__global__ __launch_bounds__(256) void gat_gemm_score_kernel(
    const float* __restrict__ msgs, const float* __restrict__ W,
    const float* __restrict__ a_src, const float* __restrict__ a_tgt,
    float* __restrict__ out,
    float* __restrict__ score_src, float* __restrict__ score_tgt)
{
    __shared__ float sA[32 * 132];

    const int r0  = blockIdx.x * 32;           // first global row of this block
    const int tid = threadIdx.x;

    // Cooperative load of the 32x128 message tile (float4, fully coalesced).
    #pragma unroll
    for (int i = 0; i < 4; ++i) {
        int idx = tid + i * 256;
        int row = idx >> 5;                    // 32 float4 per row
        int c4  = idx & 31;
        const float4 v = *reinterpret_cast<const float4*>(
            &msgs[(size_t)(r0 + row) * K_IN + c4 * 4]);
        float* dst = &sA[row * 132 + c4 * 4];
        dst[0] = v.x; dst[1] = v.y; dst[2] = v.z; dst[3] = v.w;
    }
    __syncthreads();

    const int wave = tid >> 5;                 // head index 0..7
    const int lane = tid & 31;
    const int l16  = lane & 15;
    const int half = lane >> 4;                // 0 or 1
    const int nb0  = wave * 32;                // first col of this wave's head
    const int koff = half * 2;                 // K sub-offset per WMMA A/B layout

    v8f acc00 = {0.f,0.f,0.f,0.f,0.f,0.f,0.f,0.f};
    v8f acc01 = acc00, acc10 = acc00, acc11 = acc00;

    // B fragments: B[k][n] = W[n][k]; lanes 0-15 hold cols, halves split K.
    const float* wrow0 = &W[(size_t)(nb0 +      l16) * K_IN];
    const float* wrow1 = &W[(size_t)(nb0 + 16 + l16) * K_IN];

    #pragma unroll 4
    for (int kk = 0; kk < K_IN; kk += 4) {
        v2f a0 = *reinterpret_cast<const v2f*>(&sA[(      l16) * 132 + kk + koff]);
        v2f a1 = *reinterpret_cast<const v2f*>(&sA[(16 +  l16) * 132 + kk + koff]);
        v2f b0 = *reinterpret_cast<const v2f*>(&wrow0[kk + koff]);
        v2f b1 = *reinterpret_cast<const v2f*>(&wrow1[kk + koff]);
        acc00 = __builtin_amdgcn_wmma_f32_16x16x4_f32(false, a0, false, b0,
                                                      (short)0, acc00, false, false);
        acc01 = __builtin_amdgcn_wmma_f32_16x16x4_f32(false, a0, false, b1,
                                                      (short)0, acc01, false, false);
        acc10 = __builtin_amdgcn_wmma_f32_16x16x4_f32(false, a1, false, b0,
                                                      (short)0, acc10, false, false);
        acc11 = __builtin_amdgcn_wmma_f32_16x16x4_f32(false, a1, false, b1,
                                                      (short)0, acc11, false, false);
    }

    // Epilogue. E is 16-aligned so each M tile is entirely src or entirely tgt.
    #pragma unroll
    for (int m = 0; m < 2; ++m) {
        const v8f c0 = m ? acc10 : acc00;
        const v8f c1 = m ? acc11 : acc01;
        const int rowbase = r0 + m * 16;
        const bool is_tgt = rowbase >= E_EDGES;

        if (is_tgt) {
            // proj_tgt doubles as the placeholder output rows [E, 2E).
            #pragma unroll
            for (int v = 0; v < 8; ++v) {
                int grow = rowbase + v + half * 8;       // C layout: M = v (+8 for upper half)
                out[(size_t)grow * N_OUT + nb0 +      l16] = c0[v];
                out[(size_t)grow * N_OUT + nb0 + 16 + l16] = c1[v];
            }
        }

        // score[row, head] = sum_f proj[row, head*32+f] * a[head, f]
        const float* avec = is_tgt ? a_tgt : a_src;
        const float av0 = avec[nb0 +      l16];
        const float av1 = avec[nb0 + 16 + l16];
        float* sbuf = is_tgt ? score_tgt : score_src;

        #pragma unroll
        for (int v = 0; v < 8; ++v) {
            float s = c0[v] * av0 + c1[v] * av1;
            // butterfly reduce within each 16-lane half (wave32)
            s += __shfl_xor(s, 1, 32);
            s += __shfl_xor(s, 2, 32);
            s += __shfl_xor(s, 4, 32);
            s += __shfl_xor(s, 8, 32);
            if (l16 == 0) {
                int grow = rowbase + v + half * 8;
                int e = is_tgt ? grow - E_EDGES : grow;
                sbuf[(size_t)e * H_HEADS + wave] = s;
            }
        }
    }
}

// ---------------------------------------------------------------------------
// Kernel 0: init segment buffers (d_ws is poisoned, must init every call).
// ---------------------------------------------------------------------------
__global__ void seg_init_kernel(float* __restrict__ seg_max,
                                float* __restrict__ seg_sum, int n)
{
    int i = blockIdx.x * blockDim.x + threadIdx.x;
    if (i < n) {
        seg_max[i] = -__builtin_inff();
        seg_sum[i] = 0.f;
    }
}

// ---------------------------------------------------------------------------
// Kernel 2: score = leaky_relu(score_src + score_tgt); segment max.
// (score buffer aliases score_src — read before write.)
// ---------------------------------------------------------------------------
__global__ void score_combine_kernel(float* __restrict__ score /* in: src, out: score */,
                                     const float* __restrict__ stgt,
                                     const int* __restrict__ index,
                                     float* __restrict__ seg_max, int n)
{
    int i = blockIdx.x * blockDim.x + threadIdx.x;
    if (i >= n) return;
    float s = score[i] + stgt[i];
    s = s > 0.f ? s : NEG_SLOPE * s;
    score[i] = s;
    int e = i >> 3, h = i & 7;
    atomicMax(&seg_max[(size_t)index[e] * H_HEADS + h], s);
}

// ---------------------------------------------------------------------------
// Kernel 3: ex = exp(score - seg_max[index]); segment sum. ex overwrites score.
// ---------------------------------------------------------------------------
__global__ void score_exp_kernel(float* __restrict__ score,
                                 const int* __restrict__ index,
                                 const float* __restrict__ seg_max,
                                 float* __restrict__ seg_sum, int n)
{
    int i = blockIdx.x * blockDim.x + threadIdx.x;
    if (i >= n) return;
    int e = i >> 3, h = i & 7;
    size_t sidx = (size_t)index[e] * H_HEADS + h;
    float ex = expf(score[i] - seg_max[sidx]);
    score[i] = ex;
    atomicAdd(&seg_sum[sidx], ex);
}

// ---------------------------------------------------------------------------
// Kernel 4: weighted = (ex / (denom+1e-16)) * proj_tgt, float4-vectorized.
// ---------------------------------------------------------------------------
__global__ void weighted_kernel(const float* __restrict__ ex,
                                const int* __restrict__ index,
                                const float* __restrict__ seg_sum,
                                const float* __restrict__ proj_tgt,
                                float* __restrict__ out_w, int n4)
{
    int i = blockIdx.x * blockDim.x + threadIdx.x;
    if (i >= n4) return;
    int e = i >> 6;                         // 64 float4 per edge row
    int q = i & 63;
    int h = q >> 3;                         // (q*4)/32
    float att = ex[(size_t)e * H_HEADS + h] /
                (seg_sum[(size_t)index[e] * H_HEADS + h] + 1e-16f);
    float4 p = *reinterpret_cast<const float4*>(&proj_tgt[(size_t)e * N_OUT + q * 4]);
    float4 r;
    r.x = att * p.x; r.y = att * p.y; r.z = att * p.z; r.w = att * p.w;
    *reinterpret_cast<float4*>(&out_w[(size_t)e * N_OUT + q * 4]) = r;
}

// ---------------------------------------------------------------------------
extern "C" void kernel_launch(void* const* d_in, const int* in_sizes, int n_in,
                              void* d_out, int out_size, void* d_ws, size_t ws_size,
                              hipStream_t stream)
{
    const float* msgs  = (const float*)d_in[0];   // (2E, 128)
    const float* W     = (const float*)d_in[1];   // (256, 128)
    const float* a_src = (const float*)d_in[2];   // (1, 8, 32)
    const float* a_tgt = (const float*)d_in[3];   // (1, 8, 32)
    const int*   index = (const int*)d_in[4];     // (E,)
    float* out = (float*)d_out;                   // (2E, 256)

    float* ws        = (float*)d_ws;
    float* score_src = ws;                                   // E*8  (-> score -> ex)
    float* score_tgt = score_src + (size_t)E_EDGES * H_HEADS; // E*8
    float* seg_max   = score_tgt + (size_t)E_EDGES * H_HEADS; // NSEG*8
    float* seg_sum   = seg_max   + (size_t)NSEG   * H_HEADS;  // NSEG*8

    const int nSeg   = NSEG * H_HEADS;
    const int nScore = E_EDGES * H_HEADS;

    seg_init_kernel<<<(nSeg + 255) / 256, 256, 0, stream>>>(seg_max, seg_sum, nSeg);

    gat_gemm_score_kernel<<<(2 * E_EDGES) / 32, 256, 0, stream>>>(
        msgs, W, a_src, a_tgt, out, score_src, score_tgt);

    score_combine_kernel<<<(nScore + 255) / 256, 256, 0, stream>>>(
        score_src, score_tgt, index, seg_max, nScore);

    score_exp_kernel<<<(nScore + 255) / 256, 256, 0, stream>>>(
        score_src, index, seg_max, seg_sum, nScore);

    const int n4 = E_EDGES * (N_OUT / 4);
    weighted_kernel<<<(n4 + 255) / 256, 256, 0, stream>>>(
        score_src, index, seg_sum, out + (size_t)E_EDGES * N_OUT, out, n4);
}